// AutoShiftsAug_65884798321222
// MI455X (gfx1250) — compile-verified
//
#include <hip/hip_runtime.h>
#include <hip/hip_bf16.h>

// AutoShiftsAug: per-batch constant sub-pixel translation + bilinear sample
// of a replicate-padded (PAD=4) image, zeros outside the padded extent.
// N=128, C=9, H=W=128, Hp=136. Memory-bound (~0.4 flop/byte) -> stage the
// two needed input rows per (n,y) into LDS with CDNA5 async global->LDS
// loads (ASYNCcnt path), then bilinear-combine from LDS. Output stores are
// non-temporal so the 75 MB write stream does not evict input rows from L2
// (adjacent y-blocks re-read each input row once).

#define KN 128
#define KC 9
#define KH 128
#define KW 128
#define KPAD 4
#define KHP (KH + 2 * KPAD) // 136

#define AS1 __attribute__((address_space(1)))
#define AS3 __attribute__((address_space(3)))
typedef int v4i __attribute__((ext_vector_type(4)));

#if defined(__gfx1250__) && __has_builtin(__builtin_amdgcn_global_load_async_to_lds_b128)
#define ASYNC_LDS_B128 1
#elif defined(__gfx1250__) && __has_builtin(__builtin_amdgcn_global_load_async_to_lds_b32)
#define ASYNC_LDS_B32 1
#endif

__device__ __forceinline__ void wait_async_then_barrier() {
#if defined(__gfx1250__)
#if __has_builtin(__builtin_amdgcn_s_wait_asynccnt)
    __builtin_amdgcn_s_wait_asynccnt(0);
#else
    asm volatile("s_wait_asynccnt 0" ::: "memory");
#endif
#endif
    __syncthreads();
}

__launch_bounds__(128, 8)
__global__ void AutoShiftsAug_kernel(const float* __restrict__ x,
                                     const float* __restrict__ mean,
                                     const float* __restrict__ var,
                                     const float* __restrict__ eps,
                                     const float* __restrict__ noise,
                                     float* __restrict__ out) {
    // LDS tile: 9 channels x 2 source rows x 128 cols (fp32) = 9216 B
    __shared__ float lds[2 * KC * KW];

    const int tid = threadIdx.x; // 0..127 == output x
    const int y   = blockIdx.x;  // output row
    const int n   = blockIdx.y;  // batch

    // ---- per-batch shift (all wave-uniform -> scalar loads) ----
    const float bound = 2.0f * (2 * KPAD + 1) / (float)KHP; // 18/136
    const float m0 = fminf(fmaxf(mean[0], 1e-6f), bound);
    const float m1 = fminf(fmaxf(mean[1], 1e-6f), bound);
    const float s0 = fmaxf(var[0], 1e-6f);
    const float s1 = fmaxf(var[1], 1e-6f);
    const float sh0 = fminf(fmaxf(m0 + s0 * eps[0], 0.0f), bound);
    const float sh1 = fminf(fmaxf(m1 + s1 * eps[1], 0.0f), bound);
    const float nx = noise[2 * n + 0];
    const float ny = noise[2 * n + 1];

    const float e    = 1.0f / (float)KHP;
    const float step = (2.0f - 2.0f * e) / (float)(KHP - 1);
    const float half = (float)KHP * 0.5f; // 68

    // ---- y taps for this row (wave-uniform) ----
    const float ary = -1.0f + e + step * (float)y;
    const float gy  = ((ary + sh1) + ny + 1.0f) * half - 0.5f;
    const float fy0 = floorf(gy);
    const float wy  = gy - fy0;
    const int py0   = (int)fy0;
    const bool vy0  = (py0 >= 0) && (py0 < KHP);
    const bool vy1  = (py0 + 1 >= 0) && (py0 + 1 < KHP);
    const float ay0 = vy0 ? (1.0f - wy) : 0.0f;
    const float ay1 = vy1 ? wy : 0.0f;
    const int row0  = min(max(py0 - KPAD, 0), KH - 1);     // replicate clamp
    const int row1  = min(max(py0 + 1 - KPAD, 0), KH - 1); // replicate clamp

    // ---- stage 18 rows (9 ch x 2 rows x 128 floats) into LDS ----
    const float* __restrict__ xn = x + (size_t)n * KC * KH * KW;

#if defined(ASYNC_LDS_B128)
    // 576 x b128 transfers, async global -> LDS (ASYNCcnt data path)
    for (int t = tid; t < KC * 2 * (KW / 4); t += 128) {
        const int rowid = t >> 5;        // 0..17
        const int j     = (t & 31) << 2; // col 0,4,...,124
        const int c     = rowid >> 1;
        const int srcrow = (rowid & 1) ? row1 : row0;
        const float* gsrc = xn + ((size_t)c * KH + srcrow) * KW + j;
        __builtin_amdgcn_global_load_async_to_lds_b128(
            (AS1 v4i*)gsrc, (AS3 v4i*)&lds[rowid * KW + j], 0, 0);
    }
    wait_async_then_barrier();
#elif defined(ASYNC_LDS_B32)
    for (int t = tid; t < KC * 2 * KW; t += 128) {
        const int rowid = t >> 7;      // 0..17
        const int j     = t & (KW - 1);
        const int c     = rowid >> 1;
        const int srcrow = (rowid & 1) ? row1 : row0;
        const float* gsrc = xn + ((size_t)c * KH + srcrow) * KW + j;
        __builtin_amdgcn_global_load_async_to_lds_b32(
            (AS1 int*)gsrc, (AS3 int*)&lds[rowid * KW + j], 0, 0);
    }
    wait_async_then_barrier();
#else
    // Fallback: vector loads through VGPRs + ds stores; prefetch shows the
    // gfx1250 global_prefetch_b8 path.
    for (int t = tid; t < KC * 2 * (KW / 4); t += 128) {
        const int rowid = t >> 5;
        const int j     = (t & 31) << 2;
        const int c     = rowid >> 1;
        const int srcrow = (rowid & 1) ? row1 : row0;
        const float* gsrc = xn + ((size_t)c * KH + srcrow) * KW + j;
        __builtin_prefetch(gsrc, 0, 1);
        *(float4*)&lds[rowid * KW + j] = *(const float4*)gsrc;
    }
    __syncthreads();
#endif

    // ---- x taps for this thread ----
    const float arx = -1.0f + e + step * (float)tid;
    const float gx  = ((arx + sh0) + nx + 1.0f) * half - 0.5f;
    const float fx0 = floorf(gx);
    const float wx  = gx - fx0;
    const int px0   = (int)fx0;
    const bool vx0  = (px0 >= 0) && (px0 < KHP);
    const bool vx1  = (px0 + 1 >= 0) && (px0 + 1 < KHP);
    const float ax0 = vx0 ? (1.0f - wx) : 0.0f;
    const float ax1 = vx1 ? wx : 0.0f;
    const int cc0   = min(max(px0 - KPAD, 0), KW - 1);
    const int cc1   = min(max(px0 + 1 - KPAD, 0), KW - 1);

    const float w00 = ay0 * ax0;
    const float w10 = ay0 * ax1;
    const float w01 = ay1 * ax0;
    const float w11 = ay1 * ax1;

    float* __restrict__ on = out + ((size_t)n * KC * KH + y) * KW + tid;
#pragma unroll
    for (int c = 0; c < KC; ++c) {
        const float* __restrict__ r0 = &lds[(2 * c + 0) * KW];
        const float* __restrict__ r1 = &lds[(2 * c + 1) * KW];
        const float v = w00 * r0[cc0] + w10 * r0[cc1] +
                        w01 * r1[cc0] + w11 * r1[cc1];
        // streaming write-once output: non-temporal to preserve L2 for input
        __builtin_nontemporal_store(v, on + (size_t)c * KH * KW);
    }
}

extern "C" void kernel_launch(void* const* d_in, const int* in_sizes, int n_in,
                              void* d_out, int out_size, void* d_ws, size_t ws_size,
                              hipStream_t stream) {
    (void)in_sizes; (void)n_in; (void)out_size; (void)d_ws; (void)ws_size;
    const float* x     = (const float*)d_in[0];
    const float* mean  = (const float*)d_in[1];
    const float* var   = (const float*)d_in[2];
    const float* eps   = (const float*)d_in[3];
    const float* noise = (const float*)d_in[4];
    float* out = (float*)d_out;

    dim3 grid(KH, KN); // (y, n) = (128, 128) blocks
    dim3 block(128);
    hipLaunchKernelGGL(AutoShiftsAug_kernel, grid, block, 0, stream,
                       x, mean, var, eps, noise, out);
}